// Encoder_65094524338707
// MI455X (gfx1250) — compile-verified
//
#include <hip/hip_runtime.h>
#include <stdint.h>

#define N_NODES 100000
#define N_GRAPHS 256
#define EPS_BN 1e-5f
#define NEG_SLOPE 0.2f

typedef float v2f __attribute__((ext_vector_type(2)));
typedef float v8f __attribute__((ext_vector_type(8)));

// ---------------------------------------------------------------------------
// WMMA GEMM: C[M x NCOL] = A[M x K] @ B[K x NCOL], M = 16*nTiles (exact).
// One wave32 computes one 16 x (16*NT) output tile with V_WMMA_F32_16X16X4_F32.
// A-frag layout (16x4 f32): lane m=lane&15 holds row m; VGPR0 = K=k0+(lane>>4)*2,
// VGPR1 = that +1.  B-frag (4x16): lane n=lane&15 holds col n, same K split.
// C/D (16x16 f32, 8 VGPRs): VGPR r -> row r (lanes 0-15) / r+8 (lanes 16-31).
// ---------------------------------------------------------------------------
template <int K, int NT, int NCOL, int LDA, int LDC>
__global__ void gemm_wmma(const float* __restrict__ A, const float* __restrict__ B,
                          float* __restrict__ C, int nTiles) {
    int wave = (int)((blockIdx.x * blockDim.x + threadIdx.x) >> 5);
    int lane = threadIdx.x & 31;
    if (wave >= nTiles) return;                 // wave-uniform: EXEC stays all-1 for WMMA
    const int rowBase = wave * 16;
    const int m     = lane & 15;
    const int kHalf = (lane >> 4) * 2;          // 0 or 2

    v8f acc[NT];
#pragma unroll
    for (int t = 0; t < NT; ++t) acc[t] = (v8f){0.f,0.f,0.f,0.f,0.f,0.f,0.f,0.f};

#pragma unroll
    for (int k0 = 0; k0 < K; k0 += 4) {
        v2f a;
        a.x = A[(size_t)(rowBase + m) * LDA + (k0 + kHalf)];
        a.y = A[(size_t)(rowBase + m) * LDA + (k0 + kHalf + 1)];
#pragma unroll
        for (int t = 0; t < NT; ++t) {
            const int n = t * 16 + m;
            v2f b;
            if ((NCOL % 16 == 0) || (n < NCOL)) {
                b.x = B[(size_t)(k0 + kHalf) * NCOL + n];
                b.y = B[(size_t)(k0 + kHalf + 1) * NCOL + n];
            } else {
                b.x = 0.f; b.y = 0.f;           // zero-pad columns beyond NCOL
            }
            acc[t] = __builtin_amdgcn_wmma_f32_16x16x4_f32(
                false, a, false, b, (short)0, acc[t], false, false);
        }
    }
    const int rOff = (lane >> 4) * 8;
#pragma unroll
    for (int t = 0; t < NT; ++t) {
        const int col = t * 16 + m;
#pragma unroll
        for (int r = 0; r < 8; ++r)
            C[(size_t)(rowBase + r + rOff) * LDC + col] = acc[t][r];
    }
}

// ---------------------------------------------------------------------------
// Per-node attention coefficients: as[i] = h[i,:]·a_src ; ad[i] = h[i,:]·a_dst
// ---------------------------------------------------------------------------
template <int C, int LD>
__global__ void alpha_kernel(const float* __restrict__ h,
                             const float* __restrict__ a_src,
                             const float* __restrict__ a_dst,
                             float* __restrict__ as, float* __restrict__ ad) {
    int i = blockIdx.x * blockDim.x + threadIdx.x;
    if (i >= N_NODES) return;
    const float* row = h + (size_t)i * LD;
    float s = 0.f, d = 0.f;
#pragma unroll 4
    for (int c = 0; c < C; ++c) {
        float v = row[c];
        s += v * a_src[c];
        d += v * a_dst[c];
    }
    as[i] = s; ad[i] = d;
}

// ---------- order-preserving float <-> uint (monotone): for atomic max -----
__device__ __forceinline__ unsigned ordf(float f) {
    unsigned u = __float_as_uint(f);
    return (u & 0x80000000u) ? ~u : (u | 0x80000000u);
}
__device__ __forceinline__ float unordf(unsigned o) {
    unsigned u = (o & 0x80000000u) ? (o ^ 0x80000000u) : ~o;
    return __uint_as_float(u);
}

// pass 1: e = leaky_relu(as[src]+ad[dst]); segment max over dst (ordered-uint atomicMax)
__global__ void edge_logit_max(const int* __restrict__ srcIdx, const int* __restrict__ dstIdx,
                               int E, int nTot,
                               const float* __restrict__ as, const float* __restrict__ ad,
                               float* __restrict__ eBuf, unsigned* __restrict__ emax) {
    int e = blockIdx.x * blockDim.x + threadIdx.x;
    if (e >= nTot) return;
    int s, d;
    if (e < E) { s = srcIdx[e]; d = dstIdx[e]; }
    else       { s = d = e - E; }               // self loops appended
    float v = as[s] + ad[d];
    v = (v > 0.f) ? v : v * NEG_SLOPE;
    eBuf[e] = v;
    atomicMax(&emax[d], ordf(v));
}

// pass 2: p = exp(e - emax[dst]); denom[dst] += p  (p overwrites eBuf)
__global__ void edge_exp_sum(const int* __restrict__ dstIdx, int E, int nTot,
                             const unsigned* __restrict__ emax,
                             float* __restrict__ eBuf, float* __restrict__ denom) {
    int e = blockIdx.x * blockDim.x + threadIdx.x;
    if (e >= nTot) return;
    int d = (e < E) ? dstIdx[e] : (e - E);
    float p = expf(eBuf[e] - unordf(emax[d]));
    eBuf[e] = p;
    atomicAdd(&denom[d], p);
}

// pass 3: out[dst,:] += (p/denom[dst]) * h[src,:]  — one thread per (edge, 4-chan chunk)
template <int LDH, int LDO>
__global__ void edge_aggregate(const int* __restrict__ srcIdx, const int* __restrict__ dstIdx,
                               int E, int nTot,
                               const float* __restrict__ p, const float* __restrict__ denom,
                               const float* __restrict__ h, float* __restrict__ out) {
    int e = blockIdx.x * blockDim.x + threadIdx.x;
    if (e >= nTot) return;
    int c0 = blockIdx.y * 4;
    int s, d;
    if (e < E) { s = srcIdx[e]; d = dstIdx[e]; }
    else       { s = d = e - E; }
    float coef = p[e] / denom[d];
    const float4 hv = *(const float4*)(h + (size_t)s * LDH + c0);
    float* o = out + (size_t)d * LDO + c0;
    atomicAdd(o + 0, hv.x * coef);
    atomicAdd(o + 1, hv.y * coef);
    atomicAdd(o + 2, hv.z * coef);
    atomicAdd(o + 3, hv.w * coef);
}

// BN: per-channel sum & sum-of-squares partials
template <int C, int LD>
__global__ void bn_stats(const float* __restrict__ x,
                         float* __restrict__ sums, float* __restrict__ sqs) {
    int c  = threadIdx.x % C;
    int rg = blockDim.x / C;
    int r0 = blockIdx.x * rg + threadIdx.x / C;
    int rs = gridDim.x * rg;
    float s = 0.f, q = 0.f;
    for (int r = r0; r < N_NODES; r += rs) {
        float v = x[(size_t)r * LD + c];
        s += v; q += v * v;
    }
    atomicAdd(&sums[c], s);
    atomicAdd(&sqs[c], q);
}

// BN apply in place: y = (x-mean)*rsqrt(var+eps)*g + be   (bias b cancels in BN)
template <int C, int LD>
__global__ void bn_apply(float* __restrict__ x,
                         const float* __restrict__ sums, const float* __restrict__ sqs,
                         const float* __restrict__ g, const float* __restrict__ be) {
    int i = blockIdx.x * blockDim.x + threadIdx.x;
    if (i >= N_NODES * C) return;
    int r = i / C, c = i % C;
    float mean = sums[c] * (1.0f / N_NODES);
    float var  = sqs[c]  * (1.0f / N_NODES) - mean * mean;
    float v = x[(size_t)r * LD + c];
    x[(size_t)r * LD + c] = (v - mean) * rsqrtf(var + EPS_BN) * g[c] + be[c];
}

// global mean pool (accumulate)
__global__ void pool_accum(const float* __restrict__ h, const int* __restrict__ batch,
                           float* __restrict__ psum, float* __restrict__ pcnt) {
    int i = blockIdx.x * blockDim.x + threadIdx.x;
    if (i >= N_NODES) return;
    int g = batch[i];
#pragma unroll
    for (int c = 0; c < 16; ++c)
        atomicAdd(&psum[g * 16 + c], h[(size_t)i * 16 + c]);
    atomicAdd(&pcnt[g], 1.0f);
}

__global__ void pool_final(const float* __restrict__ psum, const float* __restrict__ pcnt,
                           float* __restrict__ out) {
    int i = blockIdx.x * blockDim.x + threadIdx.x;
    if (i >= N_GRAPHS * 16) return;
    out[i] = psum[i] / fmaxf(pcnt[i / 16], 1.0f);
}

// ---------------------------------------------------------------------------
extern "C" void kernel_launch(void* const* d_in, const int* in_sizes, int n_in,
                              void* d_out, int out_size, void* d_ws, size_t ws_size,
                              hipStream_t stream) {
    (void)n_in; (void)out_size;
    const float* x      = (const float*)d_in[0];
    const int*   ei     = (const int*)  d_in[1];   // [2, E] flattened
    const int*   batch  = (const int*)  d_in[3];
    const float* W1     = (const float*)d_in[4];
    const float* a_src1 = (const float*)d_in[5];
    const float* a_dst1 = (const float*)d_in[6];
    const float* g1     = (const float*)d_in[8];
    const float* be1    = (const float*)d_in[9];
    const float* W2     = (const float*)d_in[10];
    const float* a_src2 = (const float*)d_in[11];
    const float* a_dst2 = (const float*)d_in[12];
    const float* g2     = (const float*)d_in[14];
    const float* be2    = (const float*)d_in[15];
    float* out = (float*)d_out;

    const int E    = in_sizes[1] / 2;
    const int nTot = E + N_NODES;                  // edges + self loops
    const int* src = ei;
    const int* dst = ei + E;

    // ---- workspace layout (zero-init region first, one memset) ----
    char* ws = (char*)d_ws;
    size_t off = 0;
    auto alloc = [&](size_t bytes) { size_t o = off; off = (off + bytes + 255) & ~(size_t)255; return o; };
    size_t agg1_o  = alloc((size_t)N_NODES * 64 * 4);
    size_t agg2_o  = alloc((size_t)N_NODES * 16 * 4);
    size_t den1_o  = alloc((size_t)N_NODES * 4);
    size_t emx1_o  = alloc((size_t)N_NODES * 4);
    size_t den2_o  = alloc((size_t)N_NODES * 4);
    size_t emx2_o  = alloc((size_t)N_NODES * 4);
    size_t bn1s_o  = alloc(64 * 4);
    size_t bn1q_o  = alloc(64 * 4);
    size_t bn2s_o  = alloc(64 * 4);
    size_t bn2q_o  = alloc(64 * 4);
    size_t psum_o  = alloc((size_t)N_GRAPHS * 16 * 4);
    size_t pcnt_o  = alloc((size_t)N_GRAPHS * 4);
    size_t zeroEnd = off;
    size_t h1_o    = alloc((size_t)N_NODES * 64 * 4);
    size_t h2_o    = alloc((size_t)N_NODES * 16 * 4);
    size_t as1_o   = alloc((size_t)N_NODES * 4);
    size_t ad1_o   = alloc((size_t)N_NODES * 4);
    size_t as2_o   = alloc((size_t)N_NODES * 4);
    size_t ad2_o   = alloc((size_t)N_NODES * 4);
    size_t ebuf_o  = alloc((size_t)nTot * 4);
    (void)ws_size;

    float*    agg1 = (float*)   (ws + agg1_o);
    float*    agg2 = (float*)   (ws + agg2_o);
    float*    den1 = (float*)   (ws + den1_o);
    unsigned* emx1 = (unsigned*)(ws + emx1_o);
    float*    den2 = (float*)   (ws + den2_o);
    unsigned* emx2 = (unsigned*)(ws + emx2_o);
    float*    bn1s = (float*)   (ws + bn1s_o);
    float*    bn1q = (float*)   (ws + bn1q_o);
    float*    bn2s = (float*)   (ws + bn2s_o);
    float*    bn2q = (float*)   (ws + bn2q_o);
    float*    psum = (float*)   (ws + psum_o);
    float*    pcnt = (float*)   (ws + pcnt_o);
    float*    h1   = (float*)   (ws + h1_o);
    float*    h2   = (float*)   (ws + h2_o);
    float*    as1  = (float*)   (ws + as1_o);
    float*    ad1  = (float*)   (ws + ad1_o);
    float*    as2  = (float*)   (ws + as2_o);
    float*    ad2  = (float*)   (ws + ad2_o);
    float*    eBuf = (float*)   (ws + ebuf_o);

    // zero all accumulators (memset 0 == ordered -inf for emax buffers)
    hipMemsetAsync(d_ws, 0, zeroEnd, stream);

    const int nTiles = N_NODES / 16;               // 6250 exactly
    const int gemmBlocks = (nTiles + 7) / 8;       // 8 waves/block
    const int nodeBlocks = (N_NODES + 255) / 256;
    const int edgeBlocks = (nTot + 255) / 256;

    // ---------------- layer 1 ----------------
    gemm_wmma<20, 4, 60, 20, 64><<<gemmBlocks, 256, 0, stream>>>(x, W1, h1, nTiles);
    alpha_kernel<60, 64><<<nodeBlocks, 256, 0, stream>>>(h1, a_src1, a_dst1, as1, ad1);
    edge_logit_max<<<edgeBlocks, 256, 0, stream>>>(src, dst, E, nTot, as1, ad1, eBuf, emx1);
    edge_exp_sum<<<edgeBlocks, 256, 0, stream>>>(dst, E, nTot, emx1, eBuf, den1);
    edge_aggregate<64, 64><<<dim3(edgeBlocks, 15), 256, 0, stream>>>(src, dst, E, nTot,
                                                                     eBuf, den1, h1, agg1);
    bn_stats<60, 64><<<256, 240, 0, stream>>>(agg1, bn1s, bn1q);
    bn_apply<60, 64><<<(N_NODES * 60 + 255) / 256, 256, 0, stream>>>(agg1, bn1s, bn1q, g1, be1);

    // ---------------- layer 2 ----------------
    gemm_wmma<60, 1, 16, 64, 16><<<gemmBlocks, 256, 0, stream>>>(agg1, W2, h2, nTiles);
    alpha_kernel<16, 16><<<nodeBlocks, 256, 0, stream>>>(h2, a_src2, a_dst2, as2, ad2);
    edge_logit_max<<<edgeBlocks, 256, 0, stream>>>(src, dst, E, nTot, as2, ad2, eBuf, emx2);
    edge_exp_sum<<<edgeBlocks, 256, 0, stream>>>(dst, E, nTot, emx2, eBuf, den2);
    edge_aggregate<16, 16><<<dim3(edgeBlocks, 4), 256, 0, stream>>>(src, dst, E, nTot,
                                                                    eBuf, den2, h2, agg2);
    bn_stats<16, 16><<<256, 64, 0, stream>>>(agg2, bn2s, bn2q);
    bn_apply<16, 16><<<(N_NODES * 16 + 255) / 256, 256, 0, stream>>>(agg2, bn2s, bn2q, g2, be2);

    // ---------------- pooling ----------------
    pool_accum<<<nodeBlocks, 256, 0, stream>>>(agg2, batch, psum, pcnt);
    pool_final<<<(N_GRAPHS * 16 + 255) / 256, 256, 0, stream>>>(psum, pcnt, out);
}